// SparseAttentionModule_82549271429743
// MI455X (gfx1250) — compile-verified
//
#include <hip/hip_runtime.h>
#include <hip/hip_bf16.h>
#include <math.h>

#define B_    4
#define S_    2048
#define D_    256
#define H_    8
#define HD_   32
#define E3D   (3 * D_)      // 768
#define KKEEP 204           // int(2048 * 0.1)
#define KPAD  208           // padded to multiple of 16
#define NTILES_K (KPAD / 16) // 13

typedef __attribute__((ext_vector_type(2))) float v2f;
typedef __attribute__((ext_vector_type(8))) float v8f;

static __device__ __forceinline__ v8f wmma_f32_k4(v2f a, v2f b, v8f c) {
    // D = A(16x4 f32) * B(4x16 f32) + C(16x16 f32), full fp32 precision
    return __builtin_amdgcn_wmma_f32_16x16x4_f32(false, a, false, b, (short)0, c,
                                                 false, false);
}

static __device__ __forceinline__ unsigned int float_orderable(float f) {
    unsigned int u = __float_as_uint(f);
    return (u & 0x80000000u) ? ~u : (u | 0x80000000u);
}

// ---------------------------------------------------------------------------
// Kernel 1: gate MLP -> importance[B*S]
// One wave per token; lane handles hidden units {lane, lane+32} of 64.
// ---------------------------------------------------------------------------
__global__ void sa_gate_kernel(const float* __restrict__ x,
                               const float* __restrict__ gw1,
                               const float* __restrict__ gb1,
                               const float* __restrict__ gw2,
                               const float* __restrict__ gb2,
                               float* __restrict__ imp) {
    const int wave = threadIdx.x >> 5;
    const int lane = threadIdx.x & 31;
    const int tok  = blockIdx.x * 8 + wave;
    if (tok >= B_ * S_) return;
    const float* xr = x + (size_t)tok * D_;
    float acc = 0.f;
#pragma unroll
    for (int fi = 0; fi < 2; ++fi) {
        const int f = lane + fi * 32;
        const float* wrow = gw1 + (size_t)f * D_;
        float hsum = gb1[f];
        for (int d = 0; d < D_; ++d) hsum = fmaf(xr[d], wrow[d], hsum);
        hsum = fmaxf(hsum, 0.f);           // ReLU
        acc  = fmaf(hsum, gw2[f], acc);
    }
#pragma unroll
    for (int o = 16; o >= 1; o >>= 1) acc += __shfl_xor(acc, o, 32);
    if (lane == 0) imp[tok] = 1.f / (1.f + expf(-(acc + gb2[0])));
}

// ---------------------------------------------------------------------------
// Kernel 2: per-batch top-k (k=204) radix select + index compaction.
// One block (256 threads) per batch.
// ---------------------------------------------------------------------------
__global__ void sa_topk_kernel(const float* __restrict__ imp,
                               int* __restrict__ keep_idx,
                               int* __restrict__ kcount) {
    const int b = blockIdx.x;
    const int t = threadIdx.x;
    const float* v = imp + (size_t)b * S_;
    __shared__ int hist[256];
    __shared__ unsigned int prefix;
    __shared__ int remaining;
    __shared__ int cnt;
    if (t == 0) { prefix = 0u; remaining = KKEEP; cnt = 0; }
    __syncthreads();

    for (int pass = 3; pass >= 0; --pass) {
        const int shift = pass * 8;
        hist[t] = 0;
        __syncthreads();
        const unsigned int pfx    = prefix;
        const unsigned int maskhi = (pass == 3) ? 0u : (0xFFFFFFFFu << (shift + 8));
        for (int i = t; i < S_; i += 256) {
            unsigned int u = float_orderable(v[i]);
            if ((u & maskhi) == pfx) atomicAdd(&hist[(u >> shift) & 0xFF], 1);
        }
        __syncthreads();
        if (t == 0) {
            int rem = remaining;
            unsigned int sel = 0;
            for (int bkt = 255; bkt >= 0; --bkt) {
                int c = hist[bkt];
                if (rem <= c) { sel = (unsigned int)bkt; break; }
                rem -= c;
            }
            prefix    = pfx | (sel << shift);
            remaining = rem;
        }
        __syncthreads();
    }
    const unsigned int T = prefix;  // exact bit pattern of the k-th largest
    // strictly-greater first (guaranteed < k of them), then equals up to k
    for (int i = t; i < S_; i += 256) {
        if (float_orderable(v[i]) > T) {
            int p = atomicAdd(&cnt, 1);
            keep_idx[b * KPAD + p] = i;
        }
    }
    __syncthreads();
    for (int i = t; i < S_; i += 256) {
        if (float_orderable(v[i]) == T) {
            int p = atomicAdd(&cnt, 1);
            if (p < KKEEP) keep_idx[b * KPAD + p] = i;
        }
    }
    __syncthreads();
    if (t == 0) kcount[b] = KKEEP;
    for (int i = KKEEP + t; i < KPAD; i += 256) keep_idx[b * KPAD + i] = 0; // dummy (masked)
}

// ---------------------------------------------------------------------------
// Kernel 3/5: C[M,N] = A[M,K] * W[N,K]^T + bias[N]   (fp32 WMMA 16x16x4)
// One 16x16 output tile per wave, 4 waves per block.
// ---------------------------------------------------------------------------
__global__ void sa_gemm_wt_kernel(const float* __restrict__ A,
                                  const float* __restrict__ W,
                                  const float* __restrict__ bias,
                                  float* __restrict__ C,
                                  int M, int N, int K) {
    const int wid    = blockIdx.x * (blockDim.x >> 5) + (threadIdx.x >> 5);
    const int ntiles = N >> 4;
    const int tm     = (wid / ntiles) << 4;
    const int tn     = (wid % ntiles) << 4;
    if (tm >= M) return;
    const int lane = threadIdx.x & 31;
    const int half = lane >> 4;   // 0: lanes 0-15 (K+0,K+1), 1: lanes 16-31 (K+2,K+3)
    const int l16  = lane & 15;

    const float* arow = A + (size_t)(tm + l16) * K + half * 2;
    const float* wrow = W + (size_t)(tn + l16) * K + half * 2;

    v8f c = {};
    for (int k0 = 0; k0 < K; k0 += 4) {
        v2f a, bm;
        a.x  = arow[k0];     a.y  = arow[k0 + 1];
        bm.x = wrow[k0];     bm.y = wrow[k0 + 1];
        c = wmma_f32_k4(a, bm, c);
    }
    const float bs = bias[tn + l16];
#pragma unroll
    for (int r = 0; r < 8; ++r)
        C[(size_t)(tm + r + half * 8) * N + tn + l16] = c[r] + bs;
}

// ---------------------------------------------------------------------------
// Kernel 4: sparse multi-head attention over gathered keys.
// One wave per (b, h, 16-row q tile); 4 waves / block.
// ---------------------------------------------------------------------------
__global__ void sa_attn_kernel(const float* __restrict__ qkv,  // [B*S, 768]
                               const int* __restrict__ keep_idx,
                               const int* __restrict__ kcount,
                               float* __restrict__ attn_out) { // [B*S, 256]
    __shared__ float P[4][16][KPAD];   // per-wave probability slab
    const int w    = threadIdx.x >> 5;
    const int lane = threadIdx.x & 31;
    const int half = lane >> 4;
    const int l16  = lane & 15;

    const int wid    = blockIdx.x * 4 + w;
    const int qtiles = S_ / 16;               // 128
    const int b   = wid / (H_ * qtiles);
    const int rem = wid % (H_ * qtiles);
    const int h   = rem / qtiles;
    const int qm  = (rem % qtiles) * 16;

    const int kc = kcount[b];
    const float* base = qkv + (size_t)b * S_ * E3D;
    const int* kidx   = keep_idx + b * KPAD;

    // Hoist Q A-matrix fragments: 8 k-steps covering hd=32
    v2f aq[8];
    const float* qrow = base + (size_t)(qm + l16) * E3D + h * HD_ + half * 2;
#pragma unroll
    for (int s8 = 0; s8 < 8; ++s8) { aq[s8].x = qrow[s8 * 4]; aq[s8].y = qrow[s8 * 4 + 1]; }

    const float scale = 0.17677669529663687f;  // 1/sqrt(32)

    // scores = Q * K^T over gathered keys -> LDS
    for (int nt = 0; nt < NTILES_K; ++nt) {
        const int col = nt * 16 + l16;
        const int tok = kidx[col];
        const float* kr = base + (size_t)tok * E3D + D_ + h * HD_ + half * 2;
        v8f c = {};
#pragma unroll
        for (int s8 = 0; s8 < 8; ++s8) {
            v2f bm; bm.x = kr[s8 * 4]; bm.y = kr[s8 * 4 + 1];
            c = wmma_f32_k4(aq[s8], bm, c);
        }
#pragma unroll
        for (int r = 0; r < 8; ++r) P[w][r + half * 8][col] = c[r] * scale;
    }
    __syncthreads();

    // masked softmax per row (wave-cooperative over 208 cols)
    for (int r = 0; r < 16; ++r) {
        float m = -3.4e38f;
        for (int j = lane; j < kc; j += 32) m = fmaxf(m, P[w][r][j]);
#pragma unroll
        for (int o = 16; o >= 1; o >>= 1) m = fmaxf(m, __shfl_xor(m, o, 32));
        float ssum = 0.f;
        for (int j = lane; j < KPAD; j += 32) {
            float p = (j < kc) ? expf(P[w][r][j] - m) : 0.f;
            P[w][r][j] = p;
            ssum += p;
        }
#pragma unroll
        for (int o = 16; o >= 1; o >>= 1) ssum += __shfl_xor(ssum, o, 32);
        const float inv = 1.f / ssum;
        for (int j = lane; j < KPAD; j += 32) P[w][r][j] *= inv;
    }
    __syncthreads();

    // out = P * V   (N=32 -> two 16-wide accumulators)
    v8f o0 = {}, o1 = {};
    for (int k0 = 0; k0 < KPAD; k0 += 4) {
        const int j0 = k0 + half * 2;
        v2f a; a.x = P[w][l16][j0]; a.y = P[w][l16][j0 + 1];
        const int t0 = kidx[j0], t1 = kidx[j0 + 1];
        const float* v0 = base + (size_t)t0 * E3D + 2 * D_ + h * HD_;
        const float* v1 = base + (size_t)t1 * E3D + 2 * D_ + h * HD_;
        v2f b0; b0.x = v0[l16];      b0.y = v1[l16];
        v2f b1; b1.x = v0[16 + l16]; b1.y = v1[16 + l16];
        o0 = wmma_f32_k4(a, b0, o0);
        o1 = wmma_f32_k4(a, b1, o1);
    }
#pragma unroll
    for (int r = 0; r < 8; ++r) {
        const int row = qm + r + half * 8;
        float* op = attn_out + ((size_t)b * S_ + row) * D_ + h * HD_;
        op[l16]      = o0[r];
        op[16 + l16] = o1[r];
    }
}

// ---------------------------------------------------------------------------
extern "C" void kernel_launch(void* const* d_in, const int* in_sizes, int n_in,
                              void* d_out, int out_size, void* d_ws, size_t ws_size,
                              hipStream_t stream) {
    const float* x      = (const float*)d_in[0];
    const float* w_in   = (const float*)d_in[1];
    const float* b_in   = (const float*)d_in[2];
    const float* w_out  = (const float*)d_in[3];
    const float* b_out  = (const float*)d_in[4];
    const float* gw1    = (const float*)d_in[5];
    const float* gb1    = (const float*)d_in[6];
    const float* gw2    = (const float*)d_in[7];
    const float* gb2    = (const float*)d_in[8];
    float* out = (float*)d_out;

    const int M = B_ * S_;  // 8192
    // workspace layout (floats)
    float* ws       = (float*)d_ws;
    float* qkv      = ws;                                // 8192*768
    float* attn_out = qkv + (size_t)M * E3D;             // 8192*256
    float* imp      = attn_out + (size_t)M * D_;         // 8192
    int*   keep_idx = (int*)(imp + M);                   // 4*208
    int*   kcount   = keep_idx + B_ * KPAD;              // 4

    // 1) gate MLP -> importance
    sa_gate_kernel<<<M / 8, 256, 0, stream>>>(x, gw1, gb1, gw2, gb2, imp);

    // 2) top-k per batch
    sa_topk_kernel<<<B_, 256, 0, stream>>>(imp, keep_idx, kcount);

    // 3) QKV projection: [8192,256] x [768,256]^T -> [8192,768]
    {
        const int tiles  = (M / 16) * (E3D / 16);   // 24576 waves
        sa_gemm_wt_kernel<<<tiles / 4, 128, 0, stream>>>(x, w_in, b_in, qkv,
                                                         M, E3D, D_);
    }

    // 4) sparse attention
    {
        const int waves = B_ * H_ * (S_ / 16);      // 4096
        sa_attn_kernel<<<waves / 4, 128, 0, stream>>>(qkv, keep_idx, kcount, attn_out);
    }

    // 5) output projection: [8192,256] x [256,256]^T -> d_out
    {
        const int tiles = (M / 16) * (D_ / 16);     // 8192 waves
        sa_gemm_wt_kernel<<<tiles / 4, 128, 0, stream>>>(attn_out, w_out, b_out, out,
                                                         M, D_, D_);
    }
}